// PsdQuadratic_75986561401339
// MI455X (gfx1250) — compile-verified
//
#include <hip/hip_runtime.h>

// y[i] = sum_j (x @ L)[i,j]^2 + x[i]·b + c   (B=131072, D=512, f32)
// Per-wave 16 rows; V_WMMA_F32_16X16X4_F32; two passes over N (256 cols each)
// with 16 v8f accumulators register-resident. L staged into LDS *transposed*
// (Lt[col][k], stride 20) so each WMMA B operand is one conflict-free
// ds_load_b64 directly into the operand VGPR pair. L stage copy uses
// GLOBAL_LOAD_ASYNC_TO_LDS_B32 (per-lane global->LDS scatter) + s_wait_asynccnt.

typedef float v2f __attribute__((ext_vector_type(2)));
typedef float v4f __attribute__((ext_vector_type(4)));
typedef float v8f __attribute__((ext_vector_type(8)));

#define D_DIM 512
#define ROWS_PER_WAVE 16
#define WAVES_PER_WG 4
#define ROWS_PER_WG (ROWS_PER_WAVE * WAVES_PER_WG)   // 64
#define KSTAGE 16                 // K rows of L staged per barrier epoch
#define NHALF 256                 // columns per pass
#define NT_PER_PASS (NHALF / 16)  // 16 WMMA column tiles
#define LSTRIDE 20                // 16 k + 4 pad: conflict-free ds_load_b64

#if defined(__gfx1250__) && __has_builtin(__builtin_amdgcn_global_load_async_to_lds_b32)
#define HAVE_ASYNC_LDS 1
typedef __attribute__((address_space(1))) int* gas_i32p;
typedef __attribute__((address_space(3))) int* las_i32p;
#else
#define HAVE_ASYNC_LDS 0
#endif

__global__ __launch_bounds__(128) void PsdQuadratic_75986561401339_kernel(
    const float* __restrict__ x, const float* __restrict__ L,
    const float* __restrict__ b, const float* __restrict__ c,
    float* __restrict__ out) {
  __shared__ __align__(16) float Lt[NHALF * LSTRIDE];  // 20 KB, transposed tile

  const int tid  = threadIdx.x;
  const int wave = tid >> 5;
  const int lane = tid & 31;
  const int m    = lane & 15;          // row (A) / col (B) within 16x16 tile
  const int klo  = (lane >> 4) << 1;   // 0 for lanes 0-15, 2 for lanes 16-31
  const int rowbase = blockIdx.x * ROWS_PER_WG + wave * ROWS_PER_WAVE;

  // A layout: lane m reads x[rowbase+m, k + klo .. +1] as one 8B load.
  const float* xp = x + (size_t)(rowbase + m) * D_DIM + klo;
  // B layout: lane reads Lt[(nt*16+m)*LSTRIDE + k0 + klo .. +1] as one b64.
  const int bbase = m * LSTRIDE + klo;

  v8f sq;
  #pragma unroll
  for (int r = 0; r < 8; ++r) sq[r] = 0.0f;
  float xb = 0.0f;

  for (int pass = 0; pass < 2; ++pass) {
    const int nbase = pass * NHALF;

    v8f acc[NT_PER_PASS];           // 128 VGPRs of f32 accumulators
    #pragma unroll
    for (int nt = 0; nt < NT_PER_PASS; ++nt)
      #pragma unroll
      for (int r = 0; r < 8; ++r) acc[nt][r] = 0.0f;

    for (int kstage = 0; kstage < D_DIM; kstage += KSTAGE) {
      __syncthreads();
      // Stage L[kstage..+15][nbase..+255] transposed into Lt[col][k].
      #pragma unroll
      for (int it = 0; it < 2; ++it) {
        const int colL = it * 128 + tid;                 // 0..255
        const float* gp = L + (size_t)kstage * D_DIM + nbase + colL;
#if HAVE_ASYNC_LDS
        #pragma unroll
        for (int kr = 0; kr < KSTAGE; ++kr) {
          __builtin_amdgcn_global_load_async_to_lds_b32(
              (gas_i32p)(gp + (size_t)kr * D_DIM),
              (las_i32p)&Lt[colL * LSTRIDE + kr],
              0, 0);
        }
#else
        float tmp[KSTAGE];
        #pragma unroll
        for (int kr = 0; kr < KSTAGE; ++kr)
          tmp[kr] = gp[(size_t)kr * D_DIM];
        #pragma unroll
        for (int j = 0; j < 4; ++j)
          *(v4f*)&Lt[colL * LSTRIDE + 4 * j] = *(const v4f*)&tmp[4 * j];
#endif
        if (kstage + KSTAGE < D_DIM) {
          __builtin_prefetch(gp + (size_t)KSTAGE * D_DIM, 0, 1);
        }
      }
#if HAVE_ASYNC_LDS
#if __has_builtin(__builtin_amdgcn_s_wait_asynccnt)
      __builtin_amdgcn_s_wait_asynccnt(0);
#else
      asm volatile("s_wait_asynccnt 0" ::: "memory");
#endif
#endif
      __syncthreads();

      #pragma unroll
      for (int ks4 = 0; ks4 < 4; ++ks4) {
        const int k0 = ks4 * 4;
        v2f a = *(const v2f*)(xp + kstage + k0);
        if (pass == 0) {
          v2f bb = *(const v2f*)(b + klo + kstage + k0);
          xb += a.x * bb.x + a.y * bb.y;
        }
        #pragma unroll
        for (int nt = 0; nt < NT_PER_PASS; ++nt) {
          v2f bm = *(const v2f*)&Lt[bbase + nt * (16 * LSTRIDE) + k0];
          acc[nt] = __builtin_amdgcn_wmma_f32_16x16x4_f32(
              false, a, false, bm, (short)0, acc[nt], false, false);
        }
      }
    }

    #pragma unroll
    for (int nt = 0; nt < NT_PER_PASS; ++nt)
      #pragma unroll
      for (int r = 0; r < 8; ++r) sq[r] += acc[nt][r] * acc[nt][r];
  }

  // Cross-lane reductions.
  float xbfull = xb + __shfl_xor(xb, 16, 32);
  const float cval = c[0];

  #pragma unroll
  for (int r = 0; r < 8; ++r) {
    float v = sq[r];
    v += __shfl_xor(v, 8, 32);
    v += __shfl_xor(v, 4, 32);
    v += __shfl_xor(v, 2, 32);
    v += __shfl_xor(v, 1, 32);
    // lanes 0-15 all hold row r; lanes 16-31 hold row r+8.
    if (lane == r)      out[rowbase + r]     = v + xbfull + cval;
    if (lane == r + 24) out[rowbase + r + 8] = v + xbfull + cval;
  }
}

extern "C" void kernel_launch(void* const* d_in, const int* in_sizes, int n_in,
                              void* d_out, int out_size, void* d_ws, size_t ws_size,
                              hipStream_t stream) {
  const float* x = (const float*)d_in[0];
  const float* L = (const float*)d_in[1];
  const float* b = (const float*)d_in[2];
  const float* c = (const float*)d_in[3];
  float* out = (float*)d_out;
  const int Bn = in_sizes[0] / D_DIM;          // 131072
  dim3 grid(Bn / ROWS_PER_WG), block(128);
  hipLaunchKernelGGL(PsdQuadratic_75986561401339_kernel, grid, block, 0, stream,
                     x, L, b, c, out);
}